// MemoryAccess_9534827397845
// MI455X (gfx1250) — compile-verified
//
#include <hip/hip_runtime.h>
#include <math.h>

typedef float v2f __attribute__((ext_vector_type(2)));
typedef float v4f __attribute__((ext_vector_type(4)));
typedef float v8f __attribute__((ext_vector_type(8)));

#define EPS 1e-6f

constexpr int B  = 16;
constexpr int M  = 1024;
constexpr int W  = 256;
constexpr int H  = 8;
constexpr int NW = 2;
constexpr int NR = 4;

// workspace layout (float offsets)
constexpr int WS_MEMN  = 0;                  // B*M        = 16384
constexpr int WS_KEYN  = WS_MEMN + B*M;      // B*H        = 128
constexpr int WS_SP    = WS_KEYN + B*H;      // B*H        = 128
constexpr int WS_SHARP = WS_SP   + B*H;      // B*H*M      = 131072
constexpr int WS_S     = WS_SHARP + B*H*M;   // B*NR       = 64

__device__ __forceinline__ float clamp01f(float x) { return fminf(fmaxf(x, 0.f), 1.f); }

__device__ __forceinline__ float wave_sum(float v) {
#pragma unroll
    for (int off = 16; off > 0; off >>= 1) v += __shfl_xor(v, off, 32);
    return v;
}

// ---------------- norms: L2 norms of memory rows + key rows, softplus(strengths) ----
__global__ void norms_kernel(const float* __restrict__ memory,
                             const float* __restrict__ keys,
                             const float* __restrict__ strengths,
                             float* __restrict__ ws) {
    int wid  = (blockIdx.x * blockDim.x + threadIdx.x) >> 5;
    int lane = threadIdx.x & 31;
    if (wid < B * M) {
        const float* row = memory + (size_t)wid * W;
        float s = 0.f;
#pragma unroll
        for (int i = 0; i < W / 32; ++i) { float v = row[lane + 32 * i]; s += v * v; }
        s = wave_sum(s);
        if (lane == 0) ws[WS_MEMN + wid] = sqrtf(s + EPS);
    } else {
        int r = wid - B * M;                    // key row index: b*H + h
        if (r < B * H) {
            const float* row = keys + (size_t)r * W;
            float s = 0.f;
#pragma unroll
            for (int i = 0; i < W / 32; ++i) { float v = row[lane + 32 * i]; s += v * v; }
            s = wave_sum(s);
            if (lane == 0) {
                ws[WS_KEYN + r] = sqrtf(s + EPS);
                float st = strengths[r];
                ws[WS_SP + r] = (st > 20.f) ? st : log1pf(expf(st)); // softplus
            }
        }
    }
}

// ---------------- cosine dot via V_WMMA_F32_16X16X4_F32 ----------------------------
// One wave per (batch b, m-tile of 16). A = keys (rows 8..15 zero), B = memory^T.
// sharp[b,h,m] = dot/(memN*keyN+eps) * softplus(strength) written to workspace.
__global__ void cosine_wmma_kernel(const float* __restrict__ memory,
                                   const float* __restrict__ keys,
                                   float* __restrict__ ws) {
    int wid   = (blockIdx.x * blockDim.x + threadIdx.x) >> 5;   // 0..1023
    int lane  = threadIdx.x & 31;
    int b     = wid >> 6;          // batch
    int mTile = wid & 63;          // 64 tiles of 16 along M
    int half  = lane >> 4;
    int r16   = lane & 15;
    int hrow  = r16 & 7;           // clamp key row so address is always valid
    bool validA = (r16 < H);

    const float* kbase = keys   + ((size_t)(b * H + hrow)) * W + 2 * half;
    const float* mbase = memory + ((size_t)(b * M + mTile * 16 + r16)) * W + 2 * half;

    v8f acc = {};
#pragma unroll 4
    for (int k = 0; k < W; k += 4) {
        v2f a, bf;
        float ax = kbase[k], ay = kbase[k + 1];
        a.x = validA ? ax : 0.f;
        a.y = validA ? ay : 0.f;
        bf.x = mbase[k];
        bf.y = mbase[k + 1];
        // D = A(16x4 f32) * B(4x16 f32) + C  -> v_wmma_f32_16x16x4_f32
        acc = __builtin_amdgcn_wmma_f32_16x16x4_f32(
            /*neg_a=*/false, a, /*neg_b=*/false, bf,
            /*c_mod=*/(short)0, acc, /*reuse_a=*/false, /*reuse_b=*/false);
    }

    // D layout: VGPR r, lanes 0-15 -> row M=r (h=r), col N=lane; rows 8..15 (half 1) are padding.
    if (half == 0) {
        int m = mTile * 16 + r16;
        float mn = ws[WS_MEMN + b * M + m];
#pragma unroll
        for (int r = 0; r < H; ++r) {
            float kn = ws[WS_KEYN + b * H + r];
            float sp = ws[WS_SP   + b * H + r];
            float sharp = (acc[r] / (mn * kn + EPS)) * sp;
            ws[WS_SHARP + ((size_t)(b * H + r)) * M + m] = sharp;
        }
    }
}

// ---------------- softmax over M per (b,h) row -------------------------------------
__global__ void softmax_kernel(const float* __restrict__ ws, float* __restrict__ out_cos) {
    __shared__ float red[256];
    int row = blockIdx.x;                       // b*H + h
    int t   = threadIdx.x;
    const float* x = ws + WS_SHARP + (size_t)row * M;

    float v[4];
    float vmax = -3.0e38f;
#pragma unroll
    for (int i = 0; i < 4; ++i) { v[i] = x[t + 256 * i]; vmax = fmaxf(vmax, v[i]); }
    red[t] = vmax; __syncthreads();
    for (int s = 128; s > 0; s >>= 1) { if (t < s) red[t] = fmaxf(red[t], red[t + s]); __syncthreads(); }
    float mx = red[0]; __syncthreads();

    float sum = 0.f;
#pragma unroll
    for (int i = 0; i < 4; ++i) { v[i] = expf(v[i] - mx); sum += v[i]; }
    red[t] = sum; __syncthreads();
    for (int s = 128; s > 0; s >>= 1) { if (t < s) red[t] += red[t + s]; __syncthreads(); }
    float inv = 1.f / red[0];
#pragma unroll
    for (int i = 0; i < 4; ++i) out_cos[(size_t)row * M + t + 256 * i] = v[i] * inv;
}

// ---------------- erase/write memory update (vec4 per thread) ----------------------
// memory loads stay regular-temporal (L2-resident, shared with norms/wmma kernels);
// updated_memory is written once and never re-read -> non-temporal store.
__global__ void memory_update_kernel(const float* __restrict__ memory,
                                     const float* __restrict__ write_weights,
                                     const float* __restrict__ erase_vectors,
                                     const float* __restrict__ write_vectors,
                                     float* __restrict__ out_mem) {
    int idx = blockIdx.x * blockDim.x + threadIdx.x;   // (b*M+m)*(W/4) + w4
    int w4  = idx & (W / 4 - 1);
    int bm  = idx >> 6;
    int b   = bm >> 10;
    int m   = bm & (M - 1);

    float ww0 = write_weights[(b * NW    ) * M + m];
    float ww1 = write_weights[(b * NW + 1) * M + m];

    v4f mv  = ((const v4f*)memory)[idx];
    v4f ev0 = ((const v4f*)erase_vectors)[(b * NW    ) * (W / 4) + w4];
    v4f ev1 = ((const v4f*)erase_vectors)[(b * NW + 1) * (W / 4) + w4];
    v4f wv0 = ((const v4f*)write_vectors)[(b * NW    ) * (W / 4) + w4];
    v4f wv1 = ((const v4f*)write_vectors)[(b * NW + 1) * (W / 4) + w4];

    v4f o;
#pragma unroll
    for (int c = 0; c < 4; ++c)
        o[c] = mv[c] * (1.f - clamp01f(ww0 * ev0[c] + ww1 * ev1[c]))
             + ww0 * wv0[c] + ww1 * wv1[c];
    __builtin_nontemporal_store(o, (v4f*)out_mem + idx);
}

// ---------------- usage update + precedence (= write_weights) copy -----------------
__global__ void usage_prec_kernel(const float* __restrict__ prev_usage,
                                  const float* __restrict__ write_weights,
                                  const float* __restrict__ read_weights,
                                  const float* __restrict__ free_gate,
                                  float* __restrict__ out_prec,
                                  float* __restrict__ out_usage) {
    int idx = blockIdx.x * blockDim.x + threadIdx.x;   // b*M + m
    int b = idx >> 10;
    int m = idx & (M - 1);
    float ww0 = write_weights[(b * NW    ) * M + m];
    float ww1 = write_weights[(b * NW + 1) * M + m];
    float dec = 0.f;
#pragma unroll
    for (int n = 0; n < NR; ++n)
        dec += read_weights[(b * NR + n) * M + m] * free_gate[b * NR + n];
    out_usage[idx] = clamp01f(prev_usage[idx] + ww0 + ww1 - dec);
    out_prec[(b * NW    ) * M + m] = ww0;
    out_prec[(b * NW + 1) * M + m] = ww1;
}

// ---------------- row sums of read_weights (factorized read einsum) ----------------
__global__ void rwsum_kernel(const float* __restrict__ read_weights, float* __restrict__ ws) {
    int wid  = (blockIdx.x * blockDim.x + threadIdx.x) >> 5;   // 0..B*NR-1
    int lane = threadIdx.x & 31;
    const float* row = read_weights + (size_t)wid * M;
    float s = 0.f;
#pragma unroll
    for (int i = 0; i < M / 32; ++i) s += row[lane + 32 * i];
    s = wave_sum(s);
    if (lane == 0) ws[WS_S + wid] = s;
}

__global__ void read_output_kernel(const float* __restrict__ read_vectors,
                                   const float* __restrict__ ws,
                                   float* __restrict__ out_ro) {
    int idx = blockIdx.x * blockDim.x + threadIdx.x;   // b*W + w
    int b = idx >> 8;
    int w = idx & (W - 1);
    float acc = 0.f;
#pragma unroll
    for (int n = 0; n < NR; ++n)
        acc += ws[WS_S + b * NR + n] * read_vectors[(b * NR + n) * W + w];
    out_ro[idx] = acc;
}

// ---------------- link passthrough: 128 MB streaming copy --------------------------
// 128 MB stream > half of the 192 MB L2: use non-temporal load+store so it does not
// sweep the L2 and evict the (re-read) 16 MB memory matrix.
__global__ void copy_link_kernel(const v4f* __restrict__ src,
                                 v4f* __restrict__ dst, int n4) {
    int stride = gridDim.x * blockDim.x;
    for (int i = blockIdx.x * blockDim.x + threadIdx.x; i < n4; i += stride) {
        v4f v = __builtin_nontemporal_load(src + i);
        __builtin_nontemporal_store(v, dst + i);
    }
}

extern "C" void kernel_launch(void* const* d_in, const int* in_sizes, int n_in,
                              void* d_out, int out_size, void* d_ws, size_t ws_size,
                              hipStream_t stream) {
    const float* memory        = (const float*)d_in[0];
    const float* keys          = (const float*)d_in[1];
    const float* strengths     = (const float*)d_in[2];
    const float* write_weights = (const float*)d_in[3];
    const float* free_gate     = (const float*)d_in[4];
    const float* read_weights  = (const float*)d_in[5];
    const float* prev_link     = (const float*)d_in[6];
    // d_in[7] prev_precedence_weights: unused (precedence := write_weights)
    const float* prev_usage    = (const float*)d_in[8];
    const float* erase_vectors = (const float*)d_in[9];
    const float* write_vectors = (const float*)d_in[10];
    const float* read_vectors  = (const float*)d_in[11];

    float* out       = (float*)d_out;
    float* out_mem   = out;                                           // B*M*W      = 4194304
    float* out_cos   = out + (size_t)B * M * W;                       // B*H*M      = 131072
    float* out_link  = out_cos + (size_t)B * H * M;                   // B*NW*M*M   = 33554432
    float* out_prec  = out_link + (size_t)B * NW * M * M;             // B*NW*M     = 32768
    float* out_usage = out_prec + (size_t)B * NW * M;                 // B*M        = 16384
    float* out_ro    = out_usage + (size_t)B * M;                     // B*W        = 4096

    float* ws = (float*)d_ws;

    // norms + softplus: 16384 + 128 waves
    norms_kernel<<<(B * M + B * H + 7) / 8, 256, 0, stream>>>(memory, keys, strengths, ws);

    // cosine dot via WMMA: 1024 waves (one per batch x m-tile)
    cosine_wmma_kernel<<<(B * (M / 16)) / 8, 256, 0, stream>>>(memory, keys, ws);

    // softmax over M per (b,h)
    softmax_kernel<<<B * H, 256, 0, stream>>>(ws, out_cos);

    // read-weight row sums, then factorized read output
    rwsum_kernel<<<(B * NR) / 8, 256, 0, stream>>>(read_weights, ws);
    read_output_kernel<<<(B * W) / 256, 256, 0, stream>>>(read_vectors, ws, out_ro);

    // usage + precedence
    usage_prec_kernel<<<(B * M) / 256, 256, 0, stream>>>(
        prev_usage, write_weights, read_weights, free_gate, out_prec, out_usage);

    // memory erase/write update (vec4 per thread, NT store)
    memory_update_kernel<<<(B * M * W / 4) / 256, 256, 0, stream>>>(
        memory, write_weights, erase_vectors, write_vectors, out_mem);

    // link passthrough copy (dominant bandwidth cost; fully non-temporal)
    int n4 = B * NW * M * M / 4;   // 8388608 vec4s
    copy_link_kernel<<<2048, 256, 0, stream>>>((const v4f*)prev_link, (v4f*)out_link, n4);
}